// Correlation_28303834480671
// MI455X (gfx1250) — compile-verified
//
#include <hip/hip_runtime.h>

// Cost-volume correlation (MAX_DISP=24) + 3x3 box filter for MI455X (gfx1250).
// Stage 1: WMMA f32 16x16x4 banded GEMM per (b,h) row -> ws (b,h,w,32) layout.
//          Straight-line inner loop: 8-load clause -> 3 WMMAs (no divergence).
// Stage 2: 3x3 box-sum, transposing to final (b,d,h,w) layout.

typedef __attribute__((ext_vector_type(2))) float v2f;
typedef __attribute__((ext_vector_type(8))) float v8f;

#define B_    16
#define C_    256      // channels per half (x1 / x2)
#define H_    64
#define W_    192
#define ND    25       // MAX_DISP + 1
#define NDP   32       // padded disparity stride in ws
#define CSTR  (H_ * W_)      // 12288 floats between channels
#define NTILES (W_ / 16)     // 12 w-tiles per row

__global__ __launch_bounds__(256)
void corr_wmma_kernel(const float* __restrict__ in, float* __restrict__ cws) {
    const int lane = threadIdx.x & 31;
    const int wid  = blockIdx.x * 8 + (threadIdx.x >> 5);   // one wave = one (b,h,tile)
    if (wid >= B_ * H_ * NTILES) return;                    // wave-uniform

    const int tile = wid % NTILES;
    const int bh   = wid / NTILES;
    const int h    = bh % H_;
    const int b    = bh / H_;
    const int w0   = tile * 16;
    const int half = lane >> 4;     // k-group: lanes 0-15 -> k0,k1 ; 16-31 -> k2,k3
    const int lw   = lane & 15;     // M (for A) / N (for B) index

    // x1 element (c,w) at base + c*CSTR + w ; x2 is 256 channels later.
    const size_t x1base = ((size_t)b * 512) * CSTR + (size_t)h * W_;
    const size_t coff   = (size_t)(2 * half) * CSTR;        // this lane's k channel offset
    const float* pA  = in + x1base + coff + (size_t)(w0 + lw);
    const float* pB2 = in + x1base + coff + (size_t)C_ * CSTR + (size_t)(w0 + lw); // j = i
    const float* pB1 = pB2 - 16;                                                   // j = i-1
    const float* pB0 = pB2 - 32;                                                   // j = i-2
    // Left-edge tiles read in-bounds-but-wrong B data; their accumulators are
    // discarded at store time (replaced by the zero-padding value).
    const bool v1 = (tile >= 1);
    const bool v0 = (tile >= 2);

    v8f acc0 = {}, acc1 = {}, acc2 = {};

    #pragma unroll 4
    for (int kc = 0; kc < C_; kc += 4) {
        v2f a, b0f, b1f, b2f;
        a.x   = pA[0];   a.y   = pA[CSTR];
        b2f.x = pB2[0];  b2f.y = pB2[CSTR];
        b1f.x = pB1[0];  b1f.y = pB1[CSTR];
        b0f.x = pB0[0];  b0f.y = pB0[CSTR];
        acc2 = __builtin_amdgcn_wmma_f32_16x16x4_f32(false, a, false, b2f,
                                                     (short)0, acc2, false, false);
        acc1 = __builtin_amdgcn_wmma_f32_16x16x4_f32(false, a, false, b1f,
                                                     (short)0, acc1, false, false);
        acc0 = __builtin_amdgcn_wmma_f32_16x16x4_f32(false, a, false, b0f,
                                                     (short)0, acc0, false, false);
        pA += 4 * CSTR; pB2 += 4 * CSTR; pB1 += 4 * CSTR; pB0 += 4 * CSTR;
    }

    // C/D layout: VGPR r -> M = r (lanes 0-15) / r+8 (lanes 16-31), N = lw.
    // d = wi - wj ; ownership is disjoint across the three tiles.
    const size_t obase = (((size_t)b * H_ + h) * W_) * NDP;
    #pragma unroll
    for (int r = 0; r < 8; ++r) {
        const int m = r + 8 * half;                  // wi = w0 + m
        const size_t wb = obase + (size_t)(w0 + m) * NDP;
        if (lw <= m)     cws[wb + (m - lw)]      = acc2[r];                  // d in [0, m]
        if (lw >= m - 8) cws[wb + (m + 16 - lw)] = v1 ? acc1[r] : 0.0f;      // d in [m+1, min(24,m+16)]
        if (lw >= m + 8) cws[wb + (m + 32 - lw)] = v0 ? acc0[r] : 0.0f;      // d in [m+17, 24]
    }
}

__global__ __launch_bounds__(256)
void boxfilter_kernel(const float* __restrict__ cws, float* __restrict__ out) {
    const int idx = blockIdx.x * 256 + threadIdx.x;       // flat (b,d,h,w), w on lanes
    if (idx >= B_ * ND * H_ * W_) return;
    const int w = idx % W_;
    int t = idx / W_;
    const int h = t % H_;  t /= H_;
    const int d = t % ND;
    const int b = t / ND;

    float s = 0.0f;
    #pragma unroll
    for (int dh = -1; dh <= 1; ++dh) {
        const int hh = h + dh;
        if (hh < 0 || hh >= H_) continue;
        #pragma unroll
        for (int dw = -1; dw <= 1; ++dw) {
            const int ww = w + dw;
            if (ww < 0 || ww >= W_) continue;
            s += cws[(((size_t)b * H_ + hh) * W_ + ww) * NDP + d];  // L2-resident reads
        }
    }
    out[idx] = s;   // coalesced DRAM write
}

extern "C" void kernel_launch(void* const* d_in, const int* in_sizes, int n_in,
                              void* d_out, int out_size, void* d_ws, size_t ws_size,
                              hipStream_t stream) {
    (void)in_sizes; (void)n_in; (void)out_size;
    const float* in  = (const float*)d_in[0];
    float*       out = (float*)d_out;
    float*       cws = (float*)d_ws;   // needs B_*H_*W_*NDP*4 = ~25.2 MB
    (void)ws_size;

    const int nwaves  = B_ * H_ * NTILES;                // 12288 waves
    const int blocks1 = (nwaves + 7) / 8;                // 8 waves / 256-thread block
    corr_wmma_kernel<<<blocks1, 256, 0, stream>>>(in, cws);

    const int nout    = B_ * ND * H_ * W_;               // 4,915,200
    const int blocks2 = (nout + 255) / 256;
    boxfilter_kernel<<<blocks2, 256, 0, stream>>>(cws, out);
}